// GaussianLSTM_36197984371158
// MI455X (gfx1250) — compile-verified
//
#include <hip/hip_runtime.h>
#include <math.h>

typedef __attribute__((ext_vector_type(2))) float v2f;
typedef __attribute__((ext_vector_type(8))) float v8f;

#define B_   4
#define S_   512
#define H_   256
#define G4_  1024
#define NWGT 32      // total persistent workgroups (16 per layer)

__device__ __forceinline__ v8f wmma4(v2f a, v2f b, v8f c) {
  // D = A(16x4 f32) * B(4x16 f32) + C(16x16 f32) ; exact f32 math
  return __builtin_amdgcn_wmma_f32_16x16x4_f32(false, a, false, b, (short)0, c, false, false);
}

__device__ __forceinline__ float sigmoidf_(float x) { return 1.f / (1.f + expf(-x)); }

// ---------------------------------------------------------------------------
// Zero h double-buffers (h0, h1) and grid-sync counter
// ---------------------------------------------------------------------------
__global__ void k_init(float* __restrict__ hbufs, unsigned* __restrict__ cnt) {
  for (unsigned i = threadIdx.x; i < 4096u; i += 256u) hbufs[i] = 0.f;
  if (threadIdx.x == 0) *cnt = 0u;
}

// ---------------------------------------------------------------------------
// Generic WMMA GEMM: out[r][n] = sum_k A(r,k) * W[n*256+k] + biasA[n](+biasB[n])
// rows r = (t<<2)|b, M=2048, K=256, N = ntiles*16.
// amode 0: A(r,k) = x[((r&3)*512 + (r>>2))*256 + k]   (x is (B,S,K))
// amode 1: A(r,k) = A[r*256 + k]
// 8 waves/block, one 16x16 C tile per wave.
// ---------------------------------------------------------------------------
__global__ void k_proj(const float* __restrict__ A, int amode,
                       const float* __restrict__ W,
                       const float* __restrict__ biasA, const float* __restrict__ biasB,
                       float* __restrict__ out, int ntiles, int ncols) {
  const int wave = threadIdx.x >> 5, lane = threadIdx.x & 31;
  const int tile = blockIdx.x * 8 + wave;
  const int mt = tile / ntiles, nt = tile % ntiles;
  const int lp = lane & 15, half = lane >> 4;

  const int arowidx = mt * 16 + lp;            // A fragment: M = lane&15 (both halves)
  const float* arow = A + (amode == 0 ? ((arowidx & 3) * 512 + (arowidx >> 2)) * 256
                                      : arowidx * 256);
  const float* wrow = W + (nt * 16 + lp) * 256;

  v8f acc = {0.f, 0.f, 0.f, 0.f, 0.f, 0.f, 0.f, 0.f};
#pragma unroll 8
  for (int kk = 0; kk < 64; ++kk) {
    const int off = kk * 4 + 2 * half;          // K pair for this lane half
    float2 a2 = *(const float2*)(arow + off);
    float2 b2 = *(const float2*)(wrow + off);
    v2f av; av.x = a2.x; av.y = a2.y;
    v2f bv; bv.x = b2.x; bv.y = b2.y;
    acc = wmma4(av, bv, acc);
  }

  const int n = nt * 16 + lp;
  const float bias = biasA[n] + (biasB ? biasB[n] : 0.f);
#pragma unroll
  for (int r = 0; r < 8; ++r) {
    const int orow = mt * 16 + r + 8 * half;    // C: M = vgpr + 8*(lane>=16)
    out[orow * ncols + n] = acc[r] + bias;
  }
}

// ---------------------------------------------------------------------------
// Fused 2-layer pipelined LSTM recurrence. 32 persistent WGs x 512 threads.
//   WGs  0..15 : layer 0, step t0 = u        (z0 = xw0[t0] + h0 @ W_hh0^T)
//   WGs 16..31 : layer 1, step t1 = u-1      (z1 = bias1 + [h0(t1);h1(t1-1)] @ [W_ih1|W_hh1]^T)
// WG wg owns hidden slice j in [wg*16, wg*16+16) => gate cols {j,j+256,j+512,j+768}.
// Weights live in LDS (layer0: 64 rows = 64KB+pad, layer1: 128 rows = 133KB;
// CDNA5 WGP has 320KB LDS). Cell state in registers. Hidden states double-
// buffered in global/L2 with a device-scope atomic grid barrier per step.
// 16 waves = 4 gates x 4 K-splits; per-wave WMMA chain <= 32 deep.
// ---------------------------------------------------------------------------
__global__ void k_lstm_fused(const float* __restrict__ xw0,
                             const float* __restrict__ w_hh0,
                             const float* __restrict__ w_ih1,
                             const float* __restrict__ w_hh1,
                             const float* __restrict__ b_ih1,
                             const float* __restrict__ b_hh1,
                             float* __restrict__ h0g, float* __restrict__ h1g,
                             float* __restrict__ hs1, unsigned* __restrict__ cnt) {
  __shared__ float wlds[128 * 260];    // weight rows, stride 260 -> conflict-free b64 reads
  __shared__ float hlds[8 * 260];      // staged h0 (rows 0..3) and h1 (rows 4..7)
  __shared__ float zpart[3][4][4][16]; // partials from kh=1..3 waves
  __shared__ float zbuf[4][4][16];     // full gate pre-activations

  const int tid  = threadIdx.x;                 // 0..511
  const int role = (blockIdx.x >= 16) ? 1 : 0;  // 0=layer0, 1=layer1
  const int wg   = role ? (blockIdx.x - 16) : blockIdx.x;
  const int wave = tid >> 5, lane = tid & 31;
  const int g = wave & 3, kh = wave >> 2;       // gate, K-split (0..3)
  const int lp = lane & 15, half = lane >> 4;
  const int ncol0 = g * 256 + wg * 16;

  // ---- preload weight gate-rows into LDS ----
  if (role == 0) {
    for (int e = tid; e < 64 * 256; e += 512) {
      const int rr = e >> 8, col = e & 255;
      const int grow = (rr >> 4) * 256 + wg * 16 + (rr & 15);
      wlds[rr * 260 + col] = w_hh0[grow * 256 + col];
    }
  } else {
    for (int e = tid; e < 128 * 256; e += 512) {
      const int rr = e >> 8, col = e & 255;
      const int rl = rr & 63;
      const int grow = (rl >> 4) * 256 + wg * 16 + (rl & 15);
      const float* src = (rr < 64) ? w_ih1 : w_hh1;
      wlds[rr * 260 + col] = src[grow * 256 + col];
    }
  }

  // ---- per-wave LDS base pointers + K range ----
  const float* wrow;
  const float* arow;
  int kbase, nkk;
  if (role == 0) {                    // K=256 split 4 ways over h0 / W_hh0
    wrow = &wlds[(g * 16 + lp) * 260];
    arow = &hlds[lp * 260];
    kbase = kh * 64;  nkk = 16;
  } else if (kh < 2) {                // K-half 0..255: h0 / W_ih1
    wrow = &wlds[(g * 16 + lp) * 260];
    arow = &hlds[lp * 260];
    kbase = kh * 128; nkk = 32;
  } else {                            // K-half 256..511: h1 / W_hh1
    wrow = &wlds[(64 + g * 16 + lp) * 260];
    arow = &hlds[(4 + lp) * 260];
    kbase = (kh - 2) * 128; nkk = 32;
  }

  float biasn = 0.f;
  if (role == 1 && lane < 16) biasn = b_ih1[ncol0 + lane] + b_hh1[ncol0 + lane];

  float creg = 0.f;                   // cell state (meaningful for tid<64)
  const int ub_b = tid >> 4, ub_j = tid & 15;
  unsigned expected = 0;
  __syncthreads();

  for (int u = 0; u <= S_; ++u) {
    const int active = role ? (u >= 1) : (u < S_);
    const int t = role ? (u - 1) : u;
    if (active) {
      // ---- stage hidden state(s); agent-scope loads bypass stale WGP$ ----
      for (int e = tid; e < 1024; e += 512) {
        hlds[(e >> 8) * 260 + (e & 255)] =
            __hip_atomic_load(&h0g[(u & 1) * 1024 + e],
                              __ATOMIC_RELAXED, __HIP_MEMORY_SCOPE_AGENT);
        if (role == 1)
          hlds[(4 + (e >> 8)) * 260 + (e & 255)] =
              __hip_atomic_load(&h1g[((u - 1) & 1) * 1024 + e],
                                __ATOMIC_RELAXED, __HIP_MEMORY_SCOPE_AGENT);
      }
      __syncthreads();

      // ---- gate GEMM partial: 16x16 tile, rows 0..3 valid ----
      v8f acc = {0.f, 0.f, 0.f, 0.f, 0.f, 0.f, 0.f, 0.f};
#pragma unroll 8
      for (int kk = 0; kk < nkk; ++kk) {
        const int off = kbase + kk * 4 + 2 * half;
        v2f av; av.x = 0.f; av.y = 0.f;
        if (lp < 4) { av.x = arow[off]; av.y = arow[off + 1]; }  // pad rows 4..15
        v2f bv; bv.x = wrow[off]; bv.y = wrow[off + 1];
        acc = wmma4(av, bv, acc);
      }
      if (kh >= 1 && lane < 16) {
#pragma unroll
        for (int r = 0; r < 4; ++r) zpart[kh - 1][g][r][lane] = acc[r];
      }
      __syncthreads();
      if (kh == 0 && lane < 16) {
#pragma unroll
        for (int r = 0; r < 4; ++r) {
          float z = acc[r] + zpart[0][g][r][lane] + zpart[1][g][r][lane]
                           + zpart[2][g][r][lane];
          z += (role == 0) ? xw0[(t * 4 + r) * G4_ + ncol0 + lane] : biasn;
          zbuf[g][r][lane] = z;
        }
      }
      __syncthreads();

      // ---- gate nonlinearity + state update for our (b, j) slice ----
      if (tid < 64) {
        const float iv = sigmoidf_(zbuf[0][ub_b][ub_j]);
        const float fv = sigmoidf_(zbuf[1][ub_b][ub_j]);
        const float gv = tanhf    (zbuf[2][ub_b][ub_j]);
        const float ov = sigmoidf_(zbuf[3][ub_b][ub_j]);
        creg = fv * creg + iv * gv;
        const float hnew = ov * tanhf(creg);
        const int j = wg * 16 + ub_j;
        if (role == 0) {
          __hip_atomic_store(&h0g[((u + 1) & 1) * 1024 + ub_b * 256 + j], hnew,
                             __ATOMIC_RELAXED, __HIP_MEMORY_SCOPE_AGENT);
        } else {
          __hip_atomic_store(&h1g[(u & 1) * 1024 + ub_b * 256 + j], hnew,
                             __ATOMIC_RELAXED, __HIP_MEMORY_SCOPE_AGENT);
          hs1[(t * 4 + ub_b) * 256 + j] = hnew;
        }
      }
    }

    // ---- grid barrier over all 32 WGs ----
    expected += NWGT;
    __threadfence();
    __syncthreads();
    if (tid == 0) {
      __hip_atomic_fetch_add(cnt, 1u, __ATOMIC_ACQ_REL, __HIP_MEMORY_SCOPE_AGENT);
      while (__hip_atomic_load(cnt, __ATOMIC_ACQUIRE, __HIP_MEMORY_SCOPE_AGENT) < expected) {
        __builtin_amdgcn_s_sleep(1);
      }
    }
    __syncthreads();
    __threadfence();
  }
}

// ---------------------------------------------------------------------------
// sample/mu/std_diag from stats (rows r=(t<<2)|b, 512 cols)
// ---------------------------------------------------------------------------
__global__ void k_outputs(const float* __restrict__ stats, const float* __restrict__ eps,
                          float* __restrict__ out, float* __restrict__ sd) {
  const unsigned gid = blockIdx.x * 256u + threadIdx.x;   // (b*512+t)*256 + h
  const unsigned h = gid & 255u;
  const unsigned bt = gid >> 8;
  const unsigned b = bt >> 9, t = bt & 511u;
  const unsigned r = (t << 2) | b;
  const float sraw = stats[r * 512u + h];
  const float mraw = stats[r * 512u + 256u + h];
  const float sp = (sraw > 20.f) ? sraw : log1pf(expf(sraw));  // softplus
  out[gid]            = mraw + sqrtf(sp) * eps[gid];           // sample
  out[524288u + gid]  = mraw;                                  // mu
  sd[gid] = sp;
}

// ---------------------------------------------------------------------------
// std_full: 537 MB diagonal embedding, pure float4 streaming stores
// ---------------------------------------------------------------------------
__global__ void k_fill_std(const float* __restrict__ sd, float4* __restrict__ out4) {
  const unsigned idx = blockIdx.x * 256u + threadIdx.x;  // < 33,554,432
  const unsigned seg = idx & 63u;                        // 4-col group within row
  const unsigned row = idx >> 6;                         // (b*512+t)*256 + i
  const unsigned i = row & 255u;
  float4 v = make_float4(0.f, 0.f, 0.f, 0.f);
  if ((i >> 2) == seg) {
    const float s = sd[row];
    ((float*)&v)[i & 3u] = s;
  }
  out4[idx] = v;
}

// ---------------------------------------------------------------------------
extern "C" void kernel_launch(void* const* d_in, const int* in_sizes, int n_in,
                              void* d_out, int out_size, void* d_ws, size_t ws_size,
                              hipStream_t stream) {
  (void)in_sizes; (void)n_in; (void)out_size; (void)ws_size;
  const float* x     = (const float*)d_in[0];
  const float* eps   = (const float*)d_in[1];
  const float* w_ih0 = (const float*)d_in[2];
  const float* w_hh0 = (const float*)d_in[3];
  const float* b_ih0 = (const float*)d_in[4];
  const float* b_hh0 = (const float*)d_in[5];
  const float* w_ih1 = (const float*)d_in[6];
  const float* w_hh1 = (const float*)d_in[7];
  const float* b_ih1 = (const float*)d_in[8];
  const float* b_hh1 = (const float*)d_in[9];
  const float* w_lin = (const float*)d_in[10];
  const float* b_lin = (const float*)d_in[11];
  float* out = (float*)d_out;

  float* ws = (float*)d_ws;
  float* xw     = ws;                      // 2,097,152 f  layer-0 gate pre-acts (t,b,4H)
  float* hs1    = ws + 2097152;            //   524,288 f  layer-1 output (t,b,H)
  float* stats  = ws + 2621440;            // 1,048,576 f  head output (r,512)
  float* sd     = ws + 3670016;            //   524,288 f  std_diag (b,t,h)
  float* h0g    = ws + 4194304;            //     2,048 f  h0 double-buffer
  float* h1g    = ws + 4196352;            //     2,048 f  h1 double-buffer
  unsigned* cnt = (unsigned*)(ws + 4198400);

  k_init      <<<1,      256, 0, stream>>>(h0g, cnt);   // zeros h0g+h1g (contiguous) + cnt
  k_proj      <<<1024,   256, 0, stream>>>(x, 0, w_ih0, b_ih0, b_hh0, xw, 64, 1024);
  k_lstm_fused<<<NWGT,   512, 0, stream>>>(xw, w_hh0, w_ih1, w_hh1, b_ih1, b_hh1,
                                           h0g, h1g, hs1, cnt);
  k_proj      <<<512,    256, 0, stream>>>(hs1, 1, w_lin, b_lin, nullptr, stats, 32, 512);
  k_outputs   <<<2048,   256, 0, stream>>>(stats, eps, out, sd);
  k_fill_std  <<<131072, 256, 0, stream>>>(sd, (float4*)(out + 1048576));
}